// FullTensorProduct_63178968924205
// MI455X (gfx1250) — compile-verified
//
#include <hip/hip_runtime.h>

typedef __attribute__((ext_vector_type(2))) float v2f;
typedef __attribute__((ext_vector_type(8))) float v8f;

#define MULCH 64                 // 64 channels
#define OUTW  1024               // 64*16 outputs per edge
#define WAVES_PER_BLOCK 8

// ---- compile-time permutation: flat output position -> LDS index (ch*16+col)
struct PermTable { unsigned short idx[OUTW]; };

constexpr PermTable make_perm() {
    PermTable t{};
    for (int p = 0; p < OUTW; ++p) {
        int ch = 0, col = 0;
        if      (p <  64) { ch = p;        col = 0; }                      // o_ss
        else if (p < 128) { ch = p - 64;   col = 1; }                      // o_vv0
        else if (p < 320) { int q = p-128; ch = q/3; col = 2  + q%3; }     // o_sv
        else if (p < 512) { int q = p-320; ch = q/3; col = 5  + q%3; }     // o_vs
        else if (p < 704) { int q = p-512; ch = q/3; col = 8  + q%3; }     // o_cross
        else              { int q = p-704; ch = q/5; col = 11 + q%5; }     // o_quad
        t.idx[p] = (unsigned short)(ch*16 + col);
    }
    return t;
}
__constant__ PermTable g_perm = make_perm();

struct alignas(8) U16x4 { unsigned short v[4]; };

// ---- per-edge B matrix M(x2): rows = feature (s1, v1x, v1y, v1z), cols = 16 outputs
__device__ __forceinline__ float m_entry(int row, int col,
                                         float s2, float vx, float vy, float vz) {
    constexpr float C3 = 0.5773502691896258f;   // 1/sqrt(3)
    constexpr float C2 = 0.7071067811865476f;   // 1/sqrt(2)
    constexpr float S6 = 0.4082482904638630f;   // 1/sqrt(6)
    float m = 0.0f;
    switch (col) {
      case 0:  if (row==0) m = s2;                                            break; // ss
      case 1:  if (row==1) m = vx*C3; else if (row==2) m = vy*C3;
               else if (row==3) m = vz*C3;                                    break; // vv0
      case 2:  if (row==0) m = vx;                                            break; // sv.x
      case 3:  if (row==0) m = vy;                                            break; // sv.y
      case 4:  if (row==0) m = vz;                                            break; // sv.z
      case 5:  if (row==1) m = s2;                                            break; // vs.x
      case 6:  if (row==2) m = s2;                                            break; // vs.y
      case 7:  if (row==3) m = s2;                                            break; // vs.z
      case 8:  if (row==2) m =  vz*C2; else if (row==3) m = -vy*C2;           break; // cross.x
      case 9:  if (row==3) m =  vx*C2; else if (row==1) m = -vz*C2;           break; // cross.y
      case 10: if (row==1) m =  vy*C2; else if (row==2) m = -vx*C2;           break; // cross.z
      case 11: if (row==1) m =  vy*C2; else if (row==2) m =  vx*C2;           break; // quad m=0
      case 12: if (row==2) m =  vz*C2; else if (row==3) m =  vy*C2;           break; // quad m=1
      case 13: if (row==1) m = -vx*S6; else if (row==2) m = -vy*S6;
               else if (row==3) m = 2.0f*vz*S6;                               break; // quad m=2
      case 14: if (row==1) m =  vz*C2; else if (row==3) m =  vx*C2;           break; // quad m=3
      case 15: if (row==1) m =  vx*C2; else if (row==2) m = -vy*C2;           break; // quad m=4
    }
    return m;
}

__global__ __launch_bounds__(256)
void FullTensorProduct_63178968924205_kernel(const float* __restrict__ x1,
                                             const float* __restrict__ x2,
                                             float* __restrict__ out, int N) {
    __shared__ float lds_out[WAVES_PER_BLOCK][OUTW];   // 32 KB

    const int lane = threadIdx.x & 31;
    const int wave = threadIdx.x >> 5;
    const int edge = blockIdx.x * WAVES_PER_BLOCK + wave;
    if (edge >= N) return;                              // wave-uniform: EXEC stays all-1s

    // x2 broadcast (same address per wave -> single merged request)
    const float4 X2 = *reinterpret_cast<const float4*>(x2 + (size_t)edge * 4);
    const float s2 = X2.x, vx = X2.y, vy = X2.z, vz = X2.w;

    const int col  = lane & 15;        // WMMA N (and B column)
    const int half = lane >> 4;        // 0 -> K=0,1   1 -> K=2,3

    // B operand directly in WMMA register layout (4x16, 2 VGPRs)
    v2f B;
    B.x = m_entry(half * 2 + 0, col, s2, vx, vy, vz);
    B.y = m_entry(half * 2 + 1, col, s2, vx, vy, vz);

    const float* x1row = x1 + (size_t)edge * (4 * MULCH);
    float* lw = lds_out[wave];

    // 4 tiles of 16 channels: OUT(16x16) = A(16x4) @ B(4x16), f32 WMMA
#pragma unroll
    for (int t = 0; t < 4; ++t) {
        const int ch = t * 16 + col;   // channel = WMMA row M
        // feature k: 0 -> s1[ch] = x1row[ch]; k=1..3 -> v1 = x1row[64+3ch+k-1]
        const int i0 = half ? (64 + 3 * ch + 1) : ch;
        const int i1 = half ? (64 + 3 * ch + 2) : (64 + 3 * ch);
        v2f A;
        A.x = x1row[i0];
        A.y = x1row[i1];

        v8f c = {0.f, 0.f, 0.f, 0.f, 0.f, 0.f, 0.f, 0.f};
        c = __builtin_amdgcn_wmma_f32_16x16x4_f32(
                /*neg_a=*/false, A, /*neg_b=*/false, B,
                /*c_mod=*/(short)0, c, /*reuse_a=*/false, /*reuse_b=*/false);

        // scatter D into LDS in (channel, col) layout
#pragma unroll
        for (int r = 0; r < 8; ++r) {
            const int chd = t * 16 + half * 8 + r;   // D row M = r + 8*half
            lw[chd * 16 + col] = c[r];
        }
    }

    // wave-local LDS RAW fence: DS ops are in-order per wave; this waits for
    // completion and blocks compiler reordering across it.
    asm volatile("s_wait_dscnt 0" ::: "memory");

    // fully-coalesced 4 KB row store: 8 x global_store_b128 per wave
    float* orow = out + (size_t)edge * OUTW;
#pragma unroll
    for (int it = 0; it < 8; ++it) {
        const int p = it * 128 + lane * 4;
        const U16x4 q = *reinterpret_cast<const U16x4*>(g_perm.idx + p);
        float4 v4;
        v4.x = lw[q.v[0]];
        v4.y = lw[q.v[1]];
        v4.z = lw[q.v[2]];
        v4.w = lw[q.v[3]];
        *reinterpret_cast<float4*>(orow + p) = v4;
    }
}

extern "C" void kernel_launch(void* const* d_in, const int* in_sizes, int n_in,
                              void* d_out, int out_size, void* d_ws, size_t ws_size,
                              hipStream_t stream) {
    const float* x1 = (const float*)d_in[0];
    const float* x2 = (const float*)d_in[1];
    float* out = (float*)d_out;
    const int N = in_sizes[1] / 4;                 // x2 is (N, 4)
    const int blocks = (N + WAVES_PER_BLOCK - 1) / WAVES_PER_BLOCK;
    FullTensorProduct_63178968924205_kernel<<<blocks, 256, 0, stream>>>(x1, x2, out, N);
}